// MultiBoxLoss_58712202936487
// MI455X (gfx1250) — compile-verified
//
#include <hip/hip_runtime.h>
#include <hip/hip_bf16.h>

// MultiBoxLoss for MI455X (gfx1250, wave32).
// Bottleneck: streaming conf_data (340 MB) once -> ~15us at 23.3 TB/s HBM.
// Kernel k_loss stages 128x81 conf tiles in LDS (coalesced float4), computes
// sum-of-exp via v_wmma_f32_16x16x4_f32 against a ones B-matrix (16 row sums
// per WMMA chain; XDL co-executes with v_exp_f32 TRANS ops), then lse/loss.
// Hard-negative mining replaced by exact histogram radix-select (top-K sum).

constexpr int B = 32;
constexpr int P = 32760;
constexpr int G = 32;
constexpr int C = 81;          // classes
constexpr int TILE = 128;      // priors per block in k_loss
constexpr int NBLK = (P + TILE - 1) / TILE;   // 256

typedef __attribute__((ext_vector_type(2))) float v2f;
typedef __attribute__((ext_vector_type(8))) float v8f;

__device__ __forceinline__ float smooth_l1(float d) {
    float ad = fabsf(d);
    return (ad < 1.0f) ? 0.5f * d * d : ad - 0.5f;
}

// ---------------- init: zero the argmax accumulators + output ---------------
__global__ void k_init(unsigned long long* __restrict__ packed,
                       float* __restrict__ out) {
    int t = blockIdx.x * blockDim.x + threadIdx.x;
    if (t < B * G) packed[t] = 0ull;
    if (t == 0) { out[0] = 0.f; out[1] = 0.f; }
}

// ---------------- match: IoU, best-truth per prior, best-prior per truth ----
__global__ void k_match(const float* __restrict__ priors,   // P x 4 (cx,cy,w,h)
                        const float* __restrict__ targets,  // B x G x 5
                        float* __restrict__ bto, int* __restrict__ bti,
                        unsigned long long* __restrict__ packed) {
    __shared__ float s_t[G * 4];
    __shared__ float s_area[G];
    __shared__ unsigned long long s_max[G];
    const int b = blockIdx.y;
    const int p = blockIdx.x * blockDim.x + threadIdx.x;
    if (threadIdx.x < G) {
        const float* tg = targets + ((size_t)b * G + threadIdx.x) * 5;
        float x1 = tg[0], y1 = tg[1], x2 = tg[2], y2 = tg[3];
        s_t[threadIdx.x * 4 + 0] = x1; s_t[threadIdx.x * 4 + 1] = y1;
        s_t[threadIdx.x * 4 + 2] = x2; s_t[threadIdx.x * 4 + 3] = y2;
        s_area[threadIdx.x] = (x2 - x1) * (y2 - y1);
        s_max[threadIdx.x] = 0ull;
    }
    __syncthreads();
    const bool valid = p < P;
    if (valid) {
        float4 pr = ((const float4*)priors)[p];          // cx,cy,w,h
        float bx1 = pr.x - pr.z * 0.5f, by1 = pr.y - pr.w * 0.5f;
        float bx2 = pr.x + pr.z * 0.5f, by2 = pr.y + pr.w * 0.5f;
        float areab = pr.z * pr.w;
        float best = -1.f; int bidx = 0;
        for (int g = 0; g < G; ++g) {
            float ax1 = s_t[g * 4 + 0], ay1 = s_t[g * 4 + 1];
            float ax2 = s_t[g * 4 + 2], ay2 = s_t[g * 4 + 3];
            float ix = fmaxf(fminf(ax2, bx2) - fmaxf(ax1, bx1), 0.f);
            float iy = fmaxf(fminf(ay2, by2) - fmaxf(ay1, by1), 0.f);
            float inter = ix * iy;
            float ov = inter / (s_area[g] + areab - inter);
            if (ov > best) { best = ov; bidx = g; }   // first g wins ties
            // argmax over priors: max overlap, ties -> smallest prior index
            unsigned long long key =
                ((unsigned long long)__float_as_uint(ov) << 32) |
                (unsigned int)(~(unsigned int)p);
            atomicMax(&s_max[g], key);
        }
        bto[(size_t)b * P + p] = best;
        bti[(size_t)b * P + p] = bidx;
    }
    __syncthreads();
    if (threadIdx.x < G)
        atomicMax(&packed[b * G + threadIdx.x], s_max[threadIdx.x]);
}

// ---------------- forced matching (sequential, matches fori_loop order) ----
__global__ void k_force(const unsigned long long* __restrict__ packed,
                        float* __restrict__ bto, int* __restrict__ bti) {
    const int b = blockIdx.x;
    if (threadIdx.x == 0) {
        for (int j = 0; j < G; ++j) {
            unsigned int low = (unsigned int)(packed[b * G + j] & 0xffffffffu);
            int bp = (int)(~low);
            bti[(size_t)b * P + bp] = j;     // later j overrides (fori_loop)
            bto[(size_t)b * P + bp] = 2.0f;
        }
    }
}

// ---------------- main streaming kernel: lse via WMMA + losses --------------
__global__ void k_loss(const float* __restrict__ loc,      // B x P x 4
                       const float* __restrict__ conf,     // B x P x 81
                       const float* __restrict__ priors,   // P x 4
                       const float* __restrict__ targets,  // B x G x 5
                       const float* __restrict__ bto, const int* __restrict__ bti,
                       float* __restrict__ mined,          // B x P
                       float* __restrict__ partL, int* __restrict__ partP,
                       float* __restrict__ partC) {
    __shared__ float s_tile[TILE * C];     // 41472 B
    __shared__ float s_sum[TILE];
    __shared__ float s_red[TILE];
    __shared__ float s_red2[TILE];
    __shared__ int   s_redi[TILE];
    const int b = blockIdx.y;
    const int p0 = blockIdx.x * TILE;
    const int t = threadIdx.x;             // blockDim.x == 128 (4 waves)

    // ---- stage conf tile (contiguous TILE*81 floats, 16B-aligned) ----
    const float* src = conf + ((size_t)b * P + p0) * C;
    if (p0 + TILE < P) __builtin_prefetch(src + TILE * C, 0, 3); // global_prefetch_b8
    const int nvalid = (min(TILE, P - p0)) * C;   // always a multiple of 4 here
    const int nv4 = nvalid >> 2;
    const float4* src4 = (const float4*)src;
    float4* dst4 = (float4*)s_tile;
    float m = -3.402823466e38f;
    for (int i = t; i < (TILE * C) / 4; i += blockDim.x) {
        float4 v = (i < nv4) ? src4[i] : make_float4(0.f, 0.f, 0.f, 0.f);
        dst4[i] = v;
        m = fmaxf(m, fmaxf(fmaxf(v.x, v.y), fmaxf(v.z, v.w)));
    }
    s_red[t] = m;
    __syncthreads();
    for (int s = TILE / 2; s > 0; s >>= 1) {
        if (t < s) s_red[t] = fmaxf(s_red[t], s_red[t + s]);
        __syncthreads();
    }
    const float M = s_red[0];              // any M keeps lse exact
    __syncthreads();

    // ---- sum-of-exp via v_wmma_f32_16x16x4_f32 with ones B-matrix ----
    // wave w handles 32 rows (two 16-row A groups). A layout (16x4 f32):
    // lanes 0-15: K={4k,4k+1}; lanes 16-31: K={4k+2,4k+3}; M = lane&15.
    {
        const int lane = t & 31, wav = t >> 5;
        const int mrow = lane & 15, khalf = lane >> 4;
        const int rbase = wav * 32;
        const float* rowA = &s_tile[(rbase + mrow) * C];
        const float* rowB = &s_tile[(rbase + 16 + mrow) * C];
        v2f ones; ones[0] = 1.f; ones[1] = 1.f;
        v8f accA = {}; v8f accB = {};
        #pragma unroll
        for (int k = 0; k < (C + 3) / 4; ++k) {          // 21 chains
            int c0 = 4 * k + 2 * khalf;
            v2f a;
            a[0] = (c0     < C) ? __expf(rowA[c0]     - M) : 0.f;
            a[1] = (c0 + 1 < C) ? __expf(rowA[c0 + 1] - M) : 0.f;
            accA = __builtin_amdgcn_wmma_f32_16x16x4_f32(
                false, a, false, ones, (short)0, accA, false, false);
            a[0] = (c0     < C) ? __expf(rowB[c0]     - M) : 0.f;
            a[1] = (c0 + 1 < C) ? __expf(rowB[c0 + 1] - M) : 0.f;
            accB = __builtin_amdgcn_wmma_f32_16x16x4_f32(
                false, a, false, ones, (short)0, accB, false, false);
        }
        // D layout: lane n<16 col n, VGPR v -> M=v (lanes 0-15) / M=v+8 (16-31)
        if (mrow == 0) {
            #pragma unroll
            for (int v = 0; v < 8; ++v) {
                s_sum[rbase + khalf * 8 + v]      = accA[v];
                s_sum[rbase + 16 + khalf * 8 + v] = accB[v];
            }
        }
    }
    __syncthreads();

    // ---- per-prior losses ----
    float myL = 0.f, myC2 = 0.f; int myPos = 0;
    const int p = p0 + t;
    if (p < P) {
        float ov = bto[(size_t)b * P + p];
        int g = bti[(size_t)b * P + p];
        const float* tg = targets + ((size_t)b * G + g) * 5;
        int conf_t = (ov < 0.5f) ? 0 : ((int)tg[4] + 1);
        float lse = M + __logf(s_sum[t]);
        float gathered = s_tile[t * C + conf_t];
        float lca = lse - gathered;
        bool pos = conf_t > 0;
        mined[(size_t)b * P + p] = pos ? 0.f : lca;
        if (pos) {
            myPos = 1; myC2 = lca;
            float4 pr = ((const float4*)priors)[p];
            float4 ld = ((const float4*)loc)[(size_t)b * P + p];
            float m0 = tg[0], m1 = tg[1], m2 = tg[2], m3 = tg[3];
            float gcx = ((m0 + m2) * 0.5f - pr.x) / (0.1f * pr.z);
            float gcy = ((m1 + m3) * 0.5f - pr.y) / (0.1f * pr.w);
            float gw = __logf((m2 - m0) / pr.z) / 0.2f;
            float gh = __logf((m3 - m1) / pr.w) / 0.2f;
            myL = smooth_l1(ld.x - gcx) + smooth_l1(ld.y - gcy) +
                  smooth_l1(ld.z - gw)  + smooth_l1(ld.w - gh);
        }
    }
    __syncthreads();
    s_red[t] = myL; s_red2[t] = myC2; s_redi[t] = myPos;
    __syncthreads();
    for (int s = TILE / 2; s > 0; s >>= 1) {
        if (t < s) {
            s_red[t] += s_red[t + s];
            s_red2[t] += s_red2[t + s];
            s_redi[t] += s_redi[t + s];
        }
        __syncthreads();
    }
    if (t == 0) {
        size_t o = (size_t)b * NBLK + blockIdx.x;
        partL[o] = s_red[0]; partC[o] = s_red2[0]; partP[o] = s_redi[0];
    }
}

// ---------------- deterministic per-batch reduction of block partials -------
__global__ void k_reduce(const float* __restrict__ partL,
                         const int* __restrict__ partP,
                         const float* __restrict__ partC,
                         int* __restrict__ num_pos, float* __restrict__ pos_lc,
                         float* __restrict__ lossl_b) {
    __shared__ float sL[NBLK], sC[NBLK];
    __shared__ int sP[NBLK];
    const int b = blockIdx.x, t = threadIdx.x;   // blockDim == NBLK == 256
    sL[t] = partL[(size_t)b * NBLK + t];
    sC[t] = partC[(size_t)b * NBLK + t];
    sP[t] = partP[(size_t)b * NBLK + t];
    __syncthreads();
    for (int s = NBLK / 2; s > 0; s >>= 1) {
        if (t < s) { sL[t] += sL[t + s]; sC[t] += sC[t + s]; sP[t] += sP[t + s]; }
        __syncthreads();
    }
    if (t == 0) { num_pos[b] = sP[0]; pos_lc[b] = sC[0]; lossl_b[b] = sL[0]; }
}

// ---------------- exact top-K sum via 4-pass radix-select -------------------
__global__ void k_select(const float* __restrict__ mined,
                         const int* __restrict__ num_pos,
                         float* __restrict__ neg_lc) {
    __shared__ unsigned int hist[256];
    __shared__ unsigned int s_prefix;
    __shared__ int s_rem;
    __shared__ float s_r[256];
    const int b = blockIdx.x, t = threadIdx.x;   // blockDim == 256
    const float* v = mined + (size_t)b * P;
    int K = min(3 * num_pos[b], P - 1);
    if (K == 0) K = 100;                         // NEG_HARD
    if (t == 0) { s_prefix = 0u; s_rem = K; }
    __syncthreads();
    for (int pass = 0; pass < 4; ++pass) {
        const int shift = 24 - pass * 8;
        hist[t] = 0u;
        __syncthreads();
        const unsigned int prefix = s_prefix;
        const unsigned int himask = (pass == 0) ? 0u : (0xffffffffu << (shift + 8));
        for (int i = t; i < P; i += 256) {
            unsigned int u = __float_as_uint(v[i]);   // v >= 0 -> monotonic bits
            if ((u & himask) == prefix) atomicAdd(&hist[(u >> shift) & 255], 1u);
        }
        __syncthreads();
        if (t == 0) {
            int rem = s_rem;
            int bin = 255;
            while (bin > 0) {
                int c = (int)hist[bin];
                if (rem <= c) break;
                rem -= c; --bin;
            }
            s_rem = rem;
            s_prefix = prefix | ((unsigned int)bin << shift);
        }
        __syncthreads();
    }
    const unsigned int T = s_prefix;
    const int rem = s_rem;
    float sum = 0.f;
    for (int i = t; i < P; i += 256) {
        float x = v[i];
        if (__float_as_uint(x) > T) sum += x;
    }
    s_r[t] = sum;
    __syncthreads();
    for (int s = 128; s > 0; s >>= 1) {
        if (t < s) s_r[t] += s_r[t + s];
        __syncthreads();
    }
    if (t == 0) neg_lc[b] = s_r[0] + (float)rem * __uint_as_float(T);
}

// ---------------- finalize ---------------------------------------------------
__global__ void k_final(const int* __restrict__ num_pos,
                        const float* __restrict__ pos_lc,
                        const float* __restrict__ neg_lc,
                        const float* __restrict__ lossl_b,
                        float* __restrict__ out) {
    __shared__ float sL[B], sC[B];
    __shared__ int sN[B];
    const int t = threadIdx.x;                   // blockDim == 32
    sL[t] = lossl_b[t]; sC[t] = pos_lc[t] + neg_lc[t]; sN[t] = num_pos[t];
    __syncthreads();
    for (int s = B / 2; s > 0; s >>= 1) {
        if (t < s) { sL[t] += sL[t + s]; sC[t] += sC[t + s]; sN[t] += sN[t + s]; }
        __syncthreads();
    }
    if (t == 0) {
        float N = (float)max(sN[0], 1);
        out[0] = sL[0] / N;
        out[1] = sC[0] / N;
    }
}

extern "C" void kernel_launch(void* const* d_in, const int* in_sizes, int n_in,
                              void* d_out, int out_size, void* d_ws, size_t ws_size,
                              hipStream_t stream) {
    const float* loc     = (const float*)d_in[0];  // B*P*4
    const float* conf    = (const float*)d_in[1];  // B*P*81
    const float* priors  = (const float*)d_in[2];  // P*4
    const float* targets = (const float*)d_in[3];  // B*G*5
    float* out = (float*)d_out;

    // workspace layout (~12.7 MB)
    char* w = (char*)d_ws;
    const size_t BP4 = (size_t)B * P * 4;
    unsigned long long* packed = (unsigned long long*)w;            // B*G*8
    size_t off = (size_t)B * G * 8;
    float* bto   = (float*)(w + off); off += BP4;
    int*   bti   = (int*)  (w + off); off += BP4;
    float* mined = (float*)(w + off); off += BP4;
    float* partL = (float*)(w + off); off += (size_t)B * NBLK * 4;
    int*   partP = (int*)  (w + off); off += (size_t)B * NBLK * 4;
    float* partC = (float*)(w + off); off += (size_t)B * NBLK * 4;
    int*   num_pos = (int*)  (w + off); off += B * 4;
    float* pos_lc  = (float*)(w + off); off += B * 4;
    float* neg_lc  = (float*)(w + off); off += B * 4;
    float* lossl_b = (float*)(w + off); off += B * 4;

    k_init<<<4, 256, 0, stream>>>(packed, out);
    k_match<<<dim3((P + 255) / 256, B), 256, 0, stream>>>(priors, targets, bto, bti, packed);
    k_force<<<B, 32, 0, stream>>>(packed, bto, bti);
    k_loss<<<dim3(NBLK, B), TILE, 0, stream>>>(loc, conf, priors, targets, bto, bti,
                                               mined, partL, partP, partC);
    k_reduce<<<B, NBLK, 0, stream>>>(partL, partP, partC, num_pos, pos_lc, lossl_b);
    k_select<<<B, 256, 0, stream>>>(mined, num_pos, neg_lc);
    k_final<<<1, 32, 0, stream>>>(num_pos, pos_lc, neg_lc, lossl_b, out);
}